// S4DKernel_24919400252023
// MI455X (gfx1250) — compile-verified
//
#include <hip/hip_runtime.h>
#include <hip/hip_bf16.h>

// ---------------- problem constants (match reference) ----------------
#define DMODEL 512
#define NSTATE 64
#define LMAX   2048
#define BATCH  8
#define Q      64            // chunk length
#define NCHUNK (LMAX / Q)    // 32
#define NPAD   16            // batch (8) padded to WMMA N=16

typedef float v2f __attribute__((ext_vector_type(2)));
typedef float v8f __attribute__((ext_vector_type(8)));

// ---------------- LDS partition (dynamic shared) ----------------
#define LDS_FLOATS (8256 + 4096 + 4160 + 4096 + 4096 + 4096 + 1024 + 1024 + 1024 + 5*64)

// ---------------- WMMA fragment helpers (f32 16x16x4, CDNA5 layouts) -------------
// A 16x4 (MxK), row-major lda=64 in LDS:
//   lanes 0-15: M=lane,   VGPR0=K0, VGPR1=K1 ; lanes 16-31: M=lane-16, K2/K3
// B 4x16 (KxN), row-major ldb:
//   lanes 0-15: N=lane, rows K0/K1 ; lanes 16-31: rows K2/K3
__device__ __forceinline__ void load_a_frags(const float* __restrict__ A, int mt,
                                             int lane, v2f* __restrict__ af) {
    const int am   = mt * 16 + (lane & 15);
    const int koff = (lane >> 4) << 1;        // 0 or 2
#pragma unroll
    for (int s = 0; s < 16; ++s) {
        const float* ap = A + am * 64 + s * 4 + koff;
        af[s].x = ap[0]; af[s].y = ap[1];      // ds_load_2addr / b64
    }
}

__device__ __forceinline__ void load_b_frags(const float* __restrict__ B, int ldb,
                                             int nb, int lane, v2f* __restrict__ bf) {
    const int bn   = nb + (lane & 15);
    const int koff = (lane >> 4) << 1;
#pragma unroll
    for (int s = 0; s < 16; ++s) {
        const float* bp = B + (s * 4 + koff) * ldb + bn;
        bf[s].x = bp[0]; bf[s].y = bp[ldb];
    }
}

__device__ __forceinline__ v8f wmma_accum(const v2f* __restrict__ af,
                                          const v2f* __restrict__ bf, v8f acc) {
#pragma unroll
    for (int s = 0; s < 16; ++s)
        acc = __builtin_amdgcn_wmma_f32_16x16x4_f32(false, af[s], false, bf[s],
                                                    (short)0, acc, false, false);
    return acc;
}

// A read fresh from LDS (used where A-reuse is impossible)
__device__ __forceinline__ v8f wmma_mm64(const float* __restrict__ A,
                                         const v2f* __restrict__ bf,
                                         int mt, v8f acc, int lane) {
    const int am   = mt * 16 + (lane & 15);
    const int koff = (lane >> 4) << 1;
#pragma unroll
    for (int s = 0; s < 16; ++s) {
        const float* ap = A + am * 64 + s * 4 + koff;
        v2f a; a.x = ap[0]; a.y = ap[1];
        acc = __builtin_amdgcn_wmma_f32_16x16x4_f32(false, a, false, bf[s],
                                                    (short)0, acc, false, false);
    }
    return acc;
}

// C/D tile 16x16: VGPR r -> row (mt*16 + r) for lanes 0-15, +8 for lanes 16-31
__device__ __forceinline__ void store_tile_lds(float* C, int ldc, int mt, int nb,
                                               v8f acc, int lane) {
    const int n     = nb + (lane & 15);
    const int mbase = mt * 16 + ((lane >> 4) << 3);
#pragma unroll
    for (int r = 0; r < 8; ++r) C[(mbase + r) * ldc + n] = acc[r];
}

// ---------------- kernel 1: x[b][l][d] -> xT[d][b][l] ----------------
__global__ void s4_transpose_in(const float* __restrict__ x, float* __restrict__ xT) {
    __shared__ float tile[64][65];
    const int lt = blockIdx.x, dt = blockIdx.y, b = blockIdx.z, tid = threadIdx.x;
    for (int i = tid; i < 64 * 64; i += 256) {
        const int ll = i >> 6, dd = i & 63;                      // coalesced over d
        tile[ll][dd] = x[((size_t)b * LMAX + lt * 64 + ll) * DMODEL + dt * 64 + dd];
    }
    __syncthreads();
    for (int i = tid; i < 64 * 64; i += 256) {
        const int dd = i >> 6, ll = i & 63;                      // coalesced over l
        xT[(((size_t)(dt * 64 + dd)) * BATCH + b) * LMAX + lt * 64 + ll] = tile[ll][dd];
    }
}

// ---------------- kernel 2: per-channel solve + kernel build + WMMA chunk scan --------
__global__ void __launch_bounds__(128)
s4_channel(const float* __restrict__ xT, float* __restrict__ yT,
           const float* __restrict__ L_param, const float* __restrict__ PL,
           const float* __restrict__ PR, const float* __restrict__ Bg,
           const float* __restrict__ Cg, const float* __restrict__ log_dt) {
    extern __shared__ float smem[];
    float* aug  = smem;                 // 64 x 129 (later reused as squaring ping buffer)
    float* dA   = aug  + 8256;          // dA^1, later dA^64
    float* dAT  = dA   + 4096;          // transposed dA, stride 65 (bank-conflict-free)
    float* Cm   = dAT  + 4160;          // Cm[l][i] = (C^T dA^{l+1})[i]
    float* Bm   = Cm   + 4096;          // Bm[i][j] = (dA^{63-j} dB)[i]
    float* Tm   = Bm   + 4096;          // Toeplitz of k_0..k_63
    float* Xc   = Tm   + 4096;          // 64 x 16 chunk inputs
    float* S0   = Xc   + 1024;          // state ping
    float* S1   = S0   + 1024;          // state pong
    float* dBv  = S1   + 1024;
    float* cvec = dBv  + 64;
    float* bvec = cvec + 64;
    float* kv   = bvec + 64;
    float* cC   = kv   + 64;

    const int d    = blockIdx.x;        // channel
    const int tid  = threadIdx.x;       // 0..127
    const int lane = tid & 31;          // wave32
    const int wave = tid >> 5;          // 0..3

    const float dt = __expf(log_dt[d]);

    // ---- phase 1: augmented system [I - A/2 | I + A/2 | B*dt] ----
    for (int idx = tid; idx < 64 * 64; idx += 128) {
        const int i = idx >> 6, j = idx & 63;
        float a = PL[(size_t)d * 128 + i * 2 + 0] * PR[(size_t)d * 128 + j] +
                  PL[(size_t)d * 128 + i * 2 + 1] * PR[(size_t)d * 128 + 64 + j];
        if (i == j) a -= __expf(L_param[(size_t)d * 64 + i]);
        a *= dt;
        const float diag = (i == j) ? 1.0f : 0.0f;
        aug[i * 129 + j]      = diag - 0.5f * a;
        aug[i * 129 + 64 + j] = diag + 0.5f * a;
    }
    if (tid < 64) {
        aug[tid * 129 + 128] = Bg[(size_t)d * 64 + tid] * dt;
        cC[tid]              = Cg[(size_t)d * 64 + tid];
    }
    __syncthreads();

    // ---- phase 2: Gauss-Jordan (M diagonally dominant; no pivoting) ----
    for (int k = 0; k < 64; ++k) {
        if (tid == k) {
            const float pinv = 1.0f / aug[k * 129 + k];
            for (int j = k; j < 129; ++j) aug[k * 129 + j] *= pinv;
        }
        __syncthreads();
        if (tid < 64 && tid != k) {
            const float f = aug[tid * 129 + k];
            for (int j = k; j < 129; ++j) aug[tid * 129 + j] -= f * aug[k * 129 + j];
        }
        __syncthreads();
    }
    // extract dA, dA^T (padded), dB
    for (int idx = tid; idx < 64 * 64; idx += 128) {
        const int i = idx >> 6, j = idx & 63;
        const float v = aug[i * 129 + 64 + j];
        dA[i * 64 + j]  = v;
        dAT[j * 65 + i] = v;
    }
    if (tid < 64) dBv[tid] = aug[tid * 129 + 128];
    __syncthreads();

    // ---- phase 3a: Cm[l][i] = (C^T dA^{l+1})[i] ----
    if (tid < 64) cvec[tid] = cC[tid];
    __syncthreads();
    for (int l = 0; l < Q; ++l) {
        float t = 0.0f;
        if (tid < 64)
            for (int j = 0; j < 64; ++j) t += dA[j * 64 + tid] * cvec[j]; // conflict-free
        __syncthreads();
        if (tid < 64) { cvec[tid] = t; Cm[l * 64 + tid] = t; }
        __syncthreads();
    }

    // ---- phase 3b: Bm[:,j] = dA^{63-j} dB ----
    if (tid < 64) { bvec[tid] = dBv[tid]; Bm[tid * 64 + 63] = dBv[tid]; }
    __syncthreads();
    for (int q = 1; q < Q; ++q) {
        float t = 0.0f;
        if (tid < 64)
            for (int j = 0; j < 64; ++j) t += dAT[j * 65 + tid] * bvec[j]; // conflict-free
        __syncthreads();
        if (tid < 64) { bvec[tid] = t; Bm[tid * 64 + (63 - q)] = t; }
        __syncthreads();
    }

    // ---- phase 3c: k_t = C . dA^t dB ; Toeplitz T ----
    if (tid < 64) {
        float acc = 0.0f;
        for (int i = 0; i < 64; ++i) acc += cC[i] * Bm[i * 64 + (63 - tid)];
        kv[tid] = acc;
    }
    __syncthreads();
    for (int idx = tid; idx < 64 * 64; idx += 128) {
        const int l = idx >> 6, j = idx & 63;
        Tm[idx] = (j <= l) ? kv[l - j] : 0.0f;
    }

    // ---- phase 4: dA <- dA^64 via 6 WMMA squarings (A-fragments hoisted per wave) ----
    float* src = dA;
    float* dst = aug;   // aug free now; reuse first 4096 floats
    for (int s = 0; s < 6; ++s) {
        __syncthreads();
        v2f af[16];
        load_a_frags(src, wave, lane, af);            // rows reused for all 4 nt tiles
#pragma unroll
        for (int nt = 0; nt < 4; ++nt) {
            v2f bf[16];
            load_b_frags(src, 64, nt * 16, lane, bf);
            v8f acc = {0.f, 0.f, 0.f, 0.f, 0.f, 0.f, 0.f, 0.f};
            acc = wmma_accum(af, bf, acc);
            store_tile_lds(dst, 64, wave, nt * 16, acc, lane);
        }
        float* t2 = src; src = dst; dst = t2;         // 6 swaps -> result in dA
    }
    __syncthreads();

    // ---- phase 5: chunk scan  h' = dA64 h + Bm X ;  Y = T X + Cm h ----
    for (int idx = tid; idx < Q * NPAD; idx += 128) { S0[idx] = 0.f; S1[idx] = 0.f; }
    float* Sp = S0;
    float* Sn = S1;
    const int mt = wave;                              // each wave owns a 16-row stripe
    for (int c = 0; c < NCHUNK; ++c) {
        __syncthreads();
        // stage X_c (Q x NPAD), batches in cols 0..7, coalesced over l
        for (int idx = tid; idx < Q * NPAD; idx += 128) {
            const int n = idx >> 6, l = idx & 63;
            float v = 0.0f;
            if (n < BATCH) v = xT[((size_t)d * BATCH + n) * LMAX + c * Q + l];
            Xc[l * NPAD + n] = v;
        }
        // prefetch next chunk (global_prefetch_b8)
        if (c + 1 < NCHUNK && tid < BATCH)
            __builtin_prefetch(&xT[((size_t)d * BATCH + tid) * LMAX + (c + 1) * Q], 0, 0);
        __syncthreads();

        // hoist B-fragments once per chunk: Xc used by T@X and Bm@X, Sp by Cm@h and dA@h
        v2f bfX[16], bfS[16];
        load_b_frags(Xc, NPAD, 0, lane, bfX);
        load_b_frags(Sp, NPAD, 0, lane, bfS);

        // output stripe: Y = T @ Xc + Cm @ h
        v8f y = {0.f, 0.f, 0.f, 0.f, 0.f, 0.f, 0.f, 0.f};
        y = wmma_mm64(Tm, bfX, mt, y, lane);
        y = wmma_mm64(Cm, bfS, mt, y, lane);
        {   // store to yT[d][b][l]: 8 consecutive l per lane -> b128 stores
            const int n = lane & 15;
            if (n < BATCH) {
                const int mbase = c * Q + mt * 16 + ((lane >> 4) << 3);
                float* p = yT + ((size_t)d * BATCH + n) * LMAX + mbase;
#pragma unroll
                for (int r = 0; r < 8; ++r) p[r] = y[r];
            }
        }
        // state stripe: h' = dA64 @ h + Bm @ Xc
        v8f s = {0.f, 0.f, 0.f, 0.f, 0.f, 0.f, 0.f, 0.f};
        s = wmma_mm64(dA, bfS, mt, s, lane);
        s = wmma_mm64(Bm, bfX, mt, s, lane);
        store_tile_lds(Sn, NPAD, mt, 0, s, lane);
        float* t2 = Sp; Sp = Sn; Sn = t2;
    }
}

// ---------------- kernel 3: yT -> y with D*x skip ----------------
__global__ void s4_transpose_out(const float* __restrict__ yT, const float* __restrict__ x,
                                 const float* __restrict__ Dp, float* __restrict__ y) {
    __shared__ float tile[64][65];
    const int lt = blockIdx.x, dt = blockIdx.y, b = blockIdx.z, tid = threadIdx.x;
    for (int i = tid; i < 64 * 64; i += 256) {
        const int dd = i >> 6, ll = i & 63;                      // coalesced over l
        tile[ll][dd] = yT[(((size_t)(dt * 64 + dd)) * BATCH + b) * LMAX + lt * 64 + ll];
    }
    __syncthreads();
    for (int i = tid; i < 64 * 64; i += 256) {
        const int ll = i >> 6, dd = i & 63;                      // coalesced over d
        const size_t gi = ((size_t)b * LMAX + lt * 64 + ll) * DMODEL + dt * 64 + dd;
        y[gi] = tile[ll][dd] + Dp[dt * 64 + dd] * x[gi];
    }
}

// ---------------- launch ----------------
extern "C" void kernel_launch(void* const* d_in, const int* in_sizes, int n_in,
                              void* d_out, int out_size, void* d_ws, size_t ws_size,
                              hipStream_t stream) {
    const float* x       = (const float*)d_in[0];
    const float* L_param = (const float*)d_in[1];
    const float* P_left  = (const float*)d_in[2];
    const float* P_right = (const float*)d_in[3];
    const float* Bg      = (const float*)d_in[4];
    const float* Cg      = (const float*)d_in[5];
    const float* Dg      = (const float*)d_in[6];
    const float* log_dt  = (const float*)d_in[7];
    float* y = (float*)d_out;

    float* xT = (float*)d_ws;                               // 512*8*2048 f32 = 33.5 MB
    float* yT = xT + (size_t)DMODEL * BATCH * LMAX;         // another 33.5 MB

    dim3 tgrid(LMAX / 64, DMODEL / 64, BATCH);
    s4_transpose_in<<<tgrid, 256, 0, stream>>>(x, xT);

    s4_channel<<<DMODEL, 128, LDS_FLOATS * sizeof(float), stream>>>(
        xT, yT, L_param, P_left, P_right, Bg, Cg, log_dt);

    s4_transpose_out<<<tgrid, 256, 0, stream>>>(yT, x, Dg, y);
}